// GraphNorm_76587856822962
// MI455X (gfx1250) — compile-verified
//
#include <hip/hip_runtime.h>

typedef float v2f __attribute__((ext_vector_type(2)));
typedef float v8f __attribute__((ext_vector_type(8)));

#define NUM_GRAPHS 2048
#define DIM 128
#define ROWS_PER_BLOCK 512
#define GN_EPS 1e-6f

// ---------------------------------------------------------------- zero scratch
__global__ __launch_bounds__(256) void gn_zero(float* __restrict__ p, int n) {
    int i = blockIdx.x * blockDim.x + threadIdx.x;
    const int stride = gridDim.x * blockDim.x;
    for (; i < n; i += stride) p[i] = 0.0f;
}

// ------------------------------------------------------- pass 1: WMMA reduce
// Each block: ROWS_PER_BLOCK rows x 128 cols. 8 waves, wave w owns cols
// [16w, 16w+16). Per 4-row group one V_WMMA_F32_16X16X4_F32 (A = segment
// indicator / all-ones, B = 4x16 feat tile) accumulates segment-run column
// sums; a second WMMA accumulates sums of squares. Global f32 atomics fire
// only at segment boundaries (~1 per 488 rows).
__global__ __launch_bounds__(256) void gn_reduce(const float* __restrict__ feat,
                                                 const int* __restrict__ seg,
                                                 float* __restrict__ sums,
                                                 float* __restrict__ sqs,
                                                 int N) {
    __shared__ int lseg[ROWS_PER_BLOCK];
    const int tid = threadIdx.x;
    const int rowBase = blockIdx.x * ROWS_PER_BLOCK;
    for (int i = tid; i < ROWS_PER_BLOCK; i += 256) {
        const int r = rowBase + i;
        lseg[i] = (r < N) ? seg[r] : -1;   // -1 sentinel: indicator never matches
    }
    __syncthreads();

    const int lane    = tid & 31;
    const int colBase = (tid >> 5) * 16;   // wave id * 16
    const int hl      = lane >> 4;         // half-wave: 0 or 1
    const int cl      = lane & 15;

    const v8f vzero = {};
    v2f ones; ones.x = 1.0f; ones.y = 1.0f;
    v8f csum = {};
    v8f csq  = {};
    int curSeg = __builtin_amdgcn_readfirstlane(lseg[0]);

    for (int g = 0; g < ROWS_PER_BLOCK / 4; ++g) {
        const int base = g * 4;
        const int s0 = __builtin_amdgcn_readfirstlane(lseg[base + 0]);
        const int s3 = __builtin_amdgcn_readfirstlane(lseg[base + 3]);

        // B tile (4x16 f32): VGPR0 = rows K0(lanes0-15)/K1(lanes16-31),
        //                    VGPR1 = rows K2/K3.
        const size_t rowA = (size_t)(rowBase + base + hl);

        if (s0 == curSeg && s3 == curSeg) {
            // ---- fast path: whole 4-row group belongs to curSeg (sorted ids)
            const float b0 = feat[rowA * DIM + colBase + cl];
            const float b1 = feat[(rowA + 2) * DIM + colBase + cl];
            v2f bv;  bv.x  = b0;      bv.y  = b1;
            v2f b2v; b2v.x = b0 * b0; b2v.y = b1 * b1;
            csum = __builtin_amdgcn_wmma_f32_16x16x4_f32(false, ones, false, bv,
                                                         (short)0, csum, false, false);
            csq  = __builtin_amdgcn_wmma_f32_16x16x4_f32(false, ones, false, b2v,
                                                         (short)0, csq, false, false);
            continue;
        }

        // ---- slow path: segment boundary (or tail) inside this group
        const int s1 = __builtin_amdgcn_readfirstlane(lseg[base + 1]);
        const int s2 = __builtin_amdgcn_readfirstlane(lseg[base + 2]);
        const int sA = hl ? s1 : s0;
        const int sB = hl ? s3 : s2;
        float b0 = 0.0f, b1 = 0.0f;
        if (sA >= 0) b0 = feat[rowA * DIM + colBase + cl];
        if (sB >= 0) b1 = feat[(rowA + 2) * DIM + colBase + cl];
        v2f bv;  bv.x  = b0;      bv.y  = b1;
        v2f b2v; b2v.x = b0 * b0; b2v.y = b1 * b1;

        int k = 0;
        while (k < 4) {
            const int sk = (k == 0) ? s0 : (k == 1) ? s1 : (k == 2) ? s2 : s3;
            if (sk < 0) break;                      // tail padding rows
            if (sk != curSeg) {                     // flush finished run
                if (lane < 16) {
                    atomicAdd(&sums[(size_t)curSeg * DIM + colBase + lane], csum[0]);
                    atomicAdd(&sqs [(size_t)curSeg * DIM + colBase + lane], csq[0]);
                }
                csum = vzero;
                csq  = vzero;
                curSeg = sk;
            }
            // indicator A (16x4): A[m][k] = (seg[row k] == curSeg), same for all m
            const float i0 = (s0 == curSeg) ? 1.0f : 0.0f;
            const float i1 = (s1 == curSeg) ? 1.0f : 0.0f;
            const float i2 = (s2 == curSeg) ? 1.0f : 0.0f;
            const float i3 = (s3 == curSeg) ? 1.0f : 0.0f;
            v2f a; a.x = hl ? i2 : i0; a.y = hl ? i3 : i1;

            csum = __builtin_amdgcn_wmma_f32_16x16x4_f32(false, a, false, bv,
                                                         (short)0, csum, false, false);
            csq  = __builtin_amdgcn_wmma_f32_16x16x4_f32(false, a, false, b2v,
                                                         (short)0, csq, false, false);
            // advance past the rows belonging to curSeg (ids are sorted)
            while (k < 4 &&
                   (((k == 0) ? s0 : (k == 1) ? s1 : (k == 2) ? s2 : s3) == curSeg))
                ++k;
        }
    }
    if (curSeg >= 0 && lane < 16) {
        atomicAdd(&sums[(size_t)curSeg * DIM + colBase + lane], csum[0]);
        atomicAdd(&sqs [(size_t)curSeg * DIM + colBase + lane], csq[0]);
    }
}

// --------------------------------------------- pass 2: per-segment finalize
// counts via binary search on the sorted segment_ids (no atomics), then fold
// everything into out = feat*scale + shift tables.
__global__ __launch_bounds__(128) void gn_finalize(const int* __restrict__ seg,
                                                   const float* __restrict__ sums,
                                                   const float* __restrict__ sqs,
                                                   const float* __restrict__ weight,
                                                   const float* __restrict__ bias,
                                                   const float* __restrict__ mscale,
                                                   float* __restrict__ scaleArr,
                                                   float* __restrict__ shiftArr,
                                                   int N) {
    const int s = blockIdx.x;
    const int d = threadIdx.x;

    int lo = 0, hi = N;
    while (lo < hi) { const int m = (lo + hi) >> 1; if (seg[m] < s) lo = m + 1; else hi = m; }
    const int start = lo;
    hi = N;
    while (lo < hi) { const int m = (lo + hi) >> 1; if (seg[m] < s + 1) lo = m + 1; else hi = m; }

    const float cnt = fmaxf((float)(lo - start), 1.0f);   // clamp like reference
    const float inv = 1.0f / cnt;
    const size_t idx = (size_t)s * DIM + d;
    const float mean = sums[idx] * inv;
    const float msq  = sqs[idx] * inv;
    const float ms   = mscale[d];
    // E[(x - mean*ms)^2] = E[x^2] - 2*ms*mean^2 + ms^2*mean^2
    float var = msq - 2.0f * ms * mean * mean + ms * ms * mean * mean;
    var = fmaxf(var, 0.0f);
    const float istd = 1.0f / sqrtf(var + GN_EPS);
    const float w = weight[d] * istd;
    scaleArr[idx] = w;
    shiftArr[idx] = bias[d] - mean * ms * w;
}

// ------------------------------------------------------- pass 3: streaming apply
__global__ __launch_bounds__(256) void gn_apply(const float* __restrict__ feat,
                                                const int* __restrict__ seg,
                                                const float* __restrict__ scaleArr,
                                                const float* __restrict__ shiftArr,
                                                float* __restrict__ out, int N) {
    const long long total  = (long long)N * (DIM / 4);   // float4 quanta
    const long long stride = (long long)gridDim.x * blockDim.x;
    for (long long i = (long long)blockIdx.x * blockDim.x + threadIdx.x;
         i < total; i += stride) {
        const long long row = i >> 5;          // DIM/4 == 32 float4 per row
        const int slot = (int)(i & 31);
        // i>>5 is identical for all 32 lanes of a wave (all strides are
        // multiples of 32) -> make the segment id scalar so scale/shift
        // addressing runs on the SALU.
        const int s = __builtin_amdgcn_readfirstlane(seg[row]);
        const float4 f  = ((const float4*)feat)[i];
        const float4 sc = ((const float4*)scaleArr)[(size_t)s * 32 + slot];
        const float4 sh = ((const float4*)shiftArr)[(size_t)s * 32 + slot];
        float4 o;
        o.x = fmaf(f.x, sc.x, sh.x);
        o.y = fmaf(f.y, sc.y, sh.y);
        o.z = fmaf(f.z, sc.z, sh.z);
        o.w = fmaf(f.w, sc.w, sh.w);
        ((float4*)out)[i] = o;
    }
}

extern "C" void kernel_launch(void* const* d_in, const int* in_sizes, int n_in,
                              void* d_out, int out_size, void* d_ws, size_t ws_size,
                              hipStream_t stream) {
    const float* feat   = (const float*)d_in[0];
    const int*   seg    = (const int*)d_in[1];
    const float* weight = (const float*)d_in[2];
    const float* bias   = (const float*)d_in[3];
    const float* mscale = (const float*)d_in[4];
    float* out = (float*)d_out;
    const int N = in_sizes[1];

    const size_t S = (size_t)NUM_GRAPHS * DIM;
    float* sums     = (float*)d_ws;       // [2048][128]
    float* sqs      = sums + S;           // [2048][128]
    float* scaleArr = sqs + S;            // [2048][128]
    float* shiftArr = scaleArr + S;       // [2048][128]

    gn_zero<<<512, 256, 0, stream>>>(sums, (int)(2 * S));

    const int nbRed = (N + ROWS_PER_BLOCK - 1) / ROWS_PER_BLOCK;
    gn_reduce<<<nbRed, 256, 0, stream>>>(feat, seg, sums, sqs, N);

    gn_finalize<<<NUM_GRAPHS, DIM, 0, stream>>>(seg, sums, sqs, weight, bias,
                                                mscale, scaleArr, shiftArr, N);

    gn_apply<<<8192, 256, 0, stream>>>(feat, seg, scaleArr, shiftArr, out, N);
}